// SceneNet_3435973836833
// MI455X (gfx1250) — compile-verified
//
#include <hip/hip_runtime.h>

typedef float v2f __attribute__((ext_vector_type(2)));
typedef float v8f __attribute__((ext_vector_type(8)));

#define HH   128
#define WWID 128
#define HWN  (HH * WWID)
#define CC   64
#define DD   64
#define BB   4
#define SSAMP 26
#define QSTRIDE 68   // 64 + 4 pad floats; 272B rows stay 16B-aligned

// ---------------------------------------------------------------------------
// Kernel 1: fold the 1x1 conv into the key/query decoders (all linear):
//   Wk[d][c] = sum_e key_w[d][e]*conv_w[e][c];  bk[d] = key_w[d]·conv_b + key_b[d]
// Tiny (2 x 64x64x64 MACs) — one 256-thread block, scalar.
// ---------------------------------------------------------------------------
__global__ void __launch_bounds__(256) fold_weights(
    const float* __restrict__ conv_w, const float* __restrict__ conv_b,
    const float* __restrict__ key_w,  const float* __restrict__ key_b,
    const float* __restrict__ query_w,const float* __restrict__ query_b,
    float* __restrict__ Wk, float* __restrict__ Wq,
    float* __restrict__ bk, float* __restrict__ bq)
{
    const int t = threadIdx.x;
    for (int j = 0; j < 16; ++j) {
        const int o = t * 16 + j;          // 0..4095
        const int d = o >> 6, c = o & 63;
        float sk = 0.f, sq = 0.f;
        for (int e = 0; e < CC; ++e) {
            const float cw = conv_w[e * CC + c];
            sk += key_w  [d * DD + e] * cw;
            sq += query_w[d * DD + e] * cw;
        }
        Wk[o] = sk; Wq[o] = sq;
    }
    if (t < DD) {
        float sk = key_b[t], sq = query_b[t];
        for (int e = 0; e < CC; ++e) {
            sk += key_w  [t * DD + e] * conv_b[e];
            sq += query_w[t * DD + e] * conv_b[e];
        }
        bk[t] = sk; bq[t] = sq;
    }
}

// ---------------------------------------------------------------------------
// Per-wave 16x64 tile GEMM:  out[16 nodes][64 dims] = x_tile(16x64) @ W^T
// using V_WMMA_F32_16X16X4_F32, K=64 in 16 steps of 4.
// A fragment (16x4 f32, 2 VGPRs): lane = {M=lane&15}; vgpr0 holds K=2*half,
// vgpr1 holds K=2*half+1 (half = lane>>4). B fragment (4x16) mirrors with
// N=lane&15. Accumulators: v8f, element r -> M = r + 8*half, N = lane&15.
// ---------------------------------------------------------------------------
__device__ __forceinline__ void wmma_tile_16x64(
    const float* __restrict__ xb,   // batch base of x: [C][HW]
    int n0,                         // first node of tile
    const float* __restrict__ W,    // folded weights [64][64] row-major (d,c)
    v8f acc[4])
{
    const int lane = threadIdx.x & 31;
    const int ln   = lane & 15;     // M index for A, N index for B
    const int half = lane >> 4;

    const v8f zero = {};
    acc[0] = zero; acc[1] = zero; acc[2] = zero; acc[3] = zero;

    for (int k = 0; k < CC; k += 4) {
        v2f a;
        a.x = xb[(size_t)(k + 2 * half    ) * HWN + n0 + ln];
        a.y = xb[(size_t)(k + 2 * half + 1) * HWN + n0 + ln];
#pragma unroll
        for (int dt = 0; dt < 4; ++dt) {
            const float* wp = W + (size_t)(dt * 16 + ln) * DD + k + 2 * half;
            v2f bfrag;
            bfrag.x = wp[0];
            bfrag.y = wp[1];
            acc[dt] = __builtin_amdgcn_wmma_f32_16x16x4_f32(
                false, a, false, bfrag, (short)0, acc[dt], false, false);
        }
    }
}

// ---------------------------------------------------------------------------
// Kernel 2: keys = x^T @ Wk^T + bk   (materialized: random-global gather needs it)
// One wave per 16-node tile; 4096 waves total.
// ---------------------------------------------------------------------------
__global__ void __launch_bounds__(256) keys_gemm(
    const float* __restrict__ x,
    const float* __restrict__ Wk, const float* __restrict__ bk,
    float* __restrict__ keys)
{
    const int wave = (int)((blockIdx.x * blockDim.x + threadIdx.x) >> 5);
    const int b    = wave >> 10;            // HW/16 = 1024 tiles per batch
    const int n0   = (wave & 1023) * 16;
    const float* xb = x + (size_t)b * CC * HWN;
    float* kb = keys + (size_t)b * HWN * DD;

    v8f acc[4];
    wmma_tile_16x64(xb, n0, Wk, acc);

    const int lane = threadIdx.x & 31;
    const int ln   = lane & 15;
    const int half = lane >> 4;
#pragma unroll
    for (int dt = 0; dt < 4; ++dt) {
        const float bias = bk[dt * 16 + ln];
#pragma unroll
        for (int r = 0; r < 8; ++r) {
            const int m = r + 8 * half;
            kb[(size_t)(n0 + m) * DD + dt * 16 + ln] = acc[dt][r] + bias;
        }
    }
}

// ---------------------------------------------------------------------------
// Kernel 3: fused query GEMM (WMMA -> LDS) + gather + 26-sample dot products.
// Queries never touch HBM. Keys gathers hit the 192MB L2 (keys = 16MB).
// ---------------------------------------------------------------------------
__global__ void __launch_bounds__(256) query_logits(
    const float* __restrict__ x,
    const float* __restrict__ Wq, const float* __restrict__ bq,
    const float* __restrict__ keys,
    const long long* __restrict__ rg,   // rand_global_inds int64 [B][HW][1]
    float* __restrict__ out)
{
    __shared__ float q_lds[8][16 * QSTRIDE];

    const int w    = threadIdx.x >> 5;      // wave in block
    const int lane = threadIdx.x & 31;
    const int wave = (int)((blockIdx.x * blockDim.x + threadIdx.x) >> 5);
    const int b    = wave >> 10;
    const int n0   = (wave & 1023) * 16;
    const float* xb = x + (size_t)b * CC * HWN;
    const float* kb = keys + (size_t)b * HWN * DD;

    v8f acc[4];
    wmma_tile_16x64(xb, n0, Wq, acc);

    const int ln   = lane & 15;
    const int half = lane >> 4;
#pragma unroll
    for (int dt = 0; dt < 4; ++dt) {
        const float bias = bq[dt * 16 + ln];
#pragma unroll
        for (int r = 0; r < 8; ++r) {
            const int m = r + 8 * half;
            q_lds[w][m * QSTRIDE + dt * 16 + ln] = acc[dt][r] + bias;
        }
    }
    // Same wave produced and consumes this LDS region; DS ops are in-order
    // within a wave, compiler inserts the s_wait_dscnt.

    // 16 nodes x 26 samples = 416 dots; 13 rounds of 32 lanes.
    for (int it = 0; it < 13; ++it) {
        const int t    = it * 32 + lane;
        const int nloc = t / SSAMP;
        const int s    = t - nloc * SSAMP;
        const int n    = n0 + nloc;

        int idx;
        if (s < 25) {
            const int y  = n >> 7;
            const int xx = n & 127;
            int yy = y  + (s / 5) - 2;
            int xc = xx + (s % 5) - 2;
            yy = yy < 0 ? -yy : (yy >= HH   ? 2 * HH   - 2 - yy : yy);   // np reflect
            xc = xc < 0 ? -xc : (xc >= WWID ? 2 * WWID - 2 - xc : xc);
            idx = (yy << 7) + xc;
        } else {
            idx = (int)rg[(size_t)b * HWN + n];
        }

        const float* kr = kb + (size_t)idx * DD;
        const float* qr = &q_lds[w][nloc * QSTRIDE];
        float dot = 0.f;
#pragma unroll
        for (int d = 0; d < DD; d += 4) {
            const float4 kv = *(const float4*)(kr + d);
            const float4 qv = *(const float4*)(qr + d);
            dot += kv.x * qv.x + kv.y * qv.y + kv.z * qv.z + kv.w * qv.w;
        }
        out[((size_t)b * HWN + n) * SSAMP + s] = dot * 0.125f;  // C^-0.5, C=64
    }
}

// ---------------------------------------------------------------------------
extern "C" void kernel_launch(void* const* d_in, const int* in_sizes, int n_in,
                              void* d_out, int out_size, void* d_ws, size_t ws_size,
                              hipStream_t stream)
{
    const float*     x       = (const float*)d_in[0];
    const float*     conv_w  = (const float*)d_in[1];
    const float*     conv_b  = (const float*)d_in[2];
    const float*     key_w   = (const float*)d_in[3];
    const float*     key_b   = (const float*)d_in[4];
    const float*     query_w = (const float*)d_in[5];
    const float*     query_b = (const float*)d_in[6];
    const long long* rg      = (const long long*)d_in[7];

    float* ws   = (float*)d_ws;
    float* Wk   = ws;                 // 4096
    float* Wq   = ws + 4096;          // 4096
    float* bk   = ws + 8192;          // 64
    float* bq   = ws + 8256;          // 64
    float* keys = ws + 8320;          // B*HW*D = 4,194,304 floats (16MB)

    float* out = (float*)d_out;       // [4][16384][26] f32

    fold_weights<<<1, 256, 0, stream>>>(conv_w, conv_b, key_w, key_b,
                                        query_w, query_b, Wk, Wq, bk, bq);
    // 4096 tile-waves, 8 waves per 256-thread block
    keys_gemm   <<<512, 256, 0, stream>>>(x, Wk, bk, keys);
    query_logits<<<512, 256, 0, stream>>>(x, Wq, bq, keys, rg, out);
}